// Agent_65352222376846
// MI455X (gfx1250) — compile-verified
//
#include <hip/hip_runtime.h>
#include <hip/hip_bf16.h>
#include <math.h>

// Problem constants (from reference)
#define B_     128
#define MAXN2_ 2050
#define H_     64
#define T_     32
#define N_     64
#define E_     64
#define LDP    66   // padded LDS row stride (floats), even -> 8B-aligned b64 loads

typedef __attribute__((ext_vector_type(2))) float v2f;
typedef __attribute__((ext_vector_type(8))) float v8f;

__device__ __forceinline__ v8f wmma4(v2f a, v2f b, v8f c) {
  // D = A(16x4,f32) * B(4x16,f32) + C(16x16,f32)
  return __builtin_amdgcn_wmma_f32_16x16x4_f32(
      /*neg_a=*/false, a, /*neg_b=*/false, b,
      /*c_mod=*/(short)0, c, /*reuse_a=*/false, /*reuse_b=*/false);
}

__global__ void Agent_copy_embeddings(const float* __restrict__ src,
                                      float* __restrict__ dst, int n4) {
  int i = blockIdx.x * blockDim.x + threadIdx.x;
  if (i < n4) ((float4*)dst)[i] = ((const float4*)src)[i];
}

__global__ __launch_bounds__(128) void Agent_gru_agg(
    const float* __restrict__ node_emb,   // (B, MAXN2, H)
    const float* __restrict__ ent_tab,    // (NUM_ENT, E)
    const float* __restrict__ rel_tab,    // (NUM_REL, E)
    const float* __restrict__ W_ih,       // (192, 128) row-major
    const float* __restrict__ W_hh,       // (192, 64)
    const float* __restrict__ b_ih,       // (192,)
    const float* __restrict__ b_hh,       // (192,)
    const int*   __restrict__ aim_nodes,  // (B, T)
    const int*   __restrict__ aim_ents,   // (B, T)
    const int*   __restrict__ nbr_nodes,  // (B, T, N)
    const int*   __restrict__ nbr_rels,   // (B, T, N)
    const int*   __restrict__ nbr_num,    // (B, T)
    float* __restrict__ out)              // (B, MAXN2, H) -- pre-copied
{
  __shared__ __align__(16) float ENT[E_];
  __shared__ __align__(16) float GIENT[3 * H_];       // ent part of gi (192)
  __shared__ __align__(16) float REL[N_][LDP];
  __shared__ __align__(16) float Hs[N_][LDP];
  __shared__ __align__(16) float partial[4][H_];

  const int bt   = blockIdx.x;       // 0 .. B*T-1
  const int b    = bt / T_;
  const int tid  = threadIdx.x;      // 0..127
  const int lane = tid & 31;
  const int wv   = tid >> 5;         // wave = M-tile (16 neighbors)

  const int num  = nbr_num[bt];
  const int aimN = aim_nodes[bt];
  const int aimE = aim_ents[bt];

  // Stage ent vector
  if (tid < E_) ENT[tid] = ent_tab[(size_t)aimE * E_ + tid];

  // Stage rel rows and h rows: 2 threads per neighbor row, float4 loads
  {
    const int row  = tid >> 1;
    const int half = tid & 1;
    const int nrel  = nbr_rels[bt * N_ + row];
    const int nnode = nbr_nodes[bt * N_ + row];
    const float* rsrc = rel_tab + (size_t)nrel * E_ + half * 32;
    const float* hsrc = node_emb + ((size_t)b * MAXN2_ + (size_t)nnode) * H_ + half * 32;
    #pragma unroll
    for (int j = 0; j < 32; j += 4) {
      float4 rv = *(const float4*)(rsrc + j);
      float4 hv = *(const float4*)(hsrc + j);
      const int c = half * 32 + j;
      REL[row][c + 0] = rv.x; REL[row][c + 1] = rv.y;
      REL[row][c + 2] = rv.z; REL[row][c + 3] = rv.w;
      Hs[row][c + 0] = hv.x;  Hs[row][c + 1] = hv.y;
      Hs[row][c + 2] = hv.z;  Hs[row][c + 3] = hv.w;
    }
  }
  __syncthreads();

  // Ent contribution to gi is row-invariant: a 192-vector. Compute once per
  // block on the VALU (12K MACs) instead of burning 192 broadcast-row WMMAs.
  {
    // threads 0..127 do g = tid; threads 0..63 also do g = 128 + tid
    for (int g = tid; g < 3 * H_; g += 128) {
      const float* wrow = W_ih + (size_t)g * (2 * E_);   // first 64 cols = ent part
      float s = 0.f;
      #pragma unroll
      for (int e = 0; e < E_; e += 4) {
        float4 w = *(const float4*)(wrow + e);
        s += ENT[e] * w.x + ENT[e + 1] * w.y + ENT[e + 2] * w.z + ENT[e + 3] * w.w;
      }
      GIENT[g] = s;
    }
  }
  __syncthreads();

  const int n16  = lane & 15;            // column within 16x16 tile
  const int kh   = lane >> 4;            // half selector
  const int koff = kh << 1;              // +0 or +2 in K

  #pragma unroll 1
  for (int jt = 0; jt < 4; ++jt) {       // hidden-dim tile (also gate column tile)
    v8f air = {}, aiz = {}, ain = {}, ahr = {}, ahz = {}, ahn = {};

    // W_ih rows for r/z/n gates at this column tile; only the rel half (cols 64..127)
    const float* Wr = W_ih + (size_t)(jt * 16 + n16) * (2 * E_) + E_;
    const float* Wz = Wr + (size_t)64  * (2 * E_);
    const float* Wn = Wr + (size_t)128 * (2 * E_);
    const float* Vr = W_hh + (size_t)(jt * 16 + n16) * H_;
    const float* Vz = Vr + (size_t)64  * H_;
    const float* Vn = Vr + (size_t)128 * H_;

    // Prefetch next tile's weight rows into cache (global_prefetch_b8)
    if (jt < 3) {
      __builtin_prefetch(Wr + 16 * (2 * E_), 0, 1);
      __builtin_prefetch(Wz + 16 * (2 * E_), 0, 1);
      __builtin_prefetch(Wn + 16 * (2 * E_), 0, 1);
      __builtin_prefetch(Vr + 16 * H_, 0, 1);
      __builtin_prefetch(Vz + 16 * H_, 0, 1);
      __builtin_prefetch(Vn + 16 * H_, 0, 1);
    }

    // gi rel part: A = gathered rel rows, K = 64
    #pragma unroll
    for (int k = 0; k < 64; k += 4) {
      v2f a = *(const v2f*)&REL[wv * 16 + n16][k + koff];
      air = wmma4(a, *(const v2f*)(Wr + k + koff), air);
      aiz = wmma4(a, *(const v2f*)(Wz + k + koff), aiz);
      ain = wmma4(a, *(const v2f*)(Wn + k + koff), ain);
    }
    // gh: A = gathered h rows, K = 64
    #pragma unroll
    for (int k = 0; k < 64; k += 4) {
      v2f a = *(const v2f*)&Hs[wv * 16 + n16][k + koff];
      ahr = wmma4(a, *(const v2f*)(Vr + k + koff), ahr);
      ahz = wmma4(a, *(const v2f*)(Vz + k + koff), ahz);
      ahn = wmma4(a, *(const v2f*)(Vn + k + koff), ahn);
    }

    // Fold biases + row-invariant ent contribution
    const float bir  = b_ih[jt * 16 + n16]       + GIENT[jt * 16 + n16];
    const float biz  = b_ih[64 + jt * 16 + n16]  + GIENT[64 + jt * 16 + n16];
    const float bin_ = b_ih[128 + jt * 16 + n16] + GIENT[128 + jt * 16 + n16];
    const float bhr  = b_hh[jt * 16 + n16];
    const float bhz  = b_hh[64 + jt * 16 + n16];
    const float bhn  = b_hh[128 + jt * 16 + n16];

    // GRU elementwise on the C/D fragment layout + masked row reduction
    float s = 0.f;
    #pragma unroll
    for (int r = 0; r < 8; ++r) {
      const int m = wv * 16 + r + (kh << 3);          // neighbor index
      const float gr = (air[r] + bir) + (ahr[r] + bhr);
      const float gz = (aiz[r] + biz) + (ahz[r] + bhz);
      const float rg = 1.f / (1.f + expf(-gr));
      const float zg = 1.f / (1.f + expf(-gz));
      const float ng = tanhf((ain[r] + bin_) + rg * (ahn[r] + bhn));
      const float hv = Hs[m][jt * 16 + n16];
      const float hnew = (1.f - zg) * ng + zg * hv;
      s += (m < num) ? hnew : 0.f;
    }
    // combine the two 8-row halves (lanes L and L^16 hold the same column)
    float tot = s + __shfl_xor(s, 16, 32);
    if (lane < 16) partial[wv][jt * 16 + lane] = tot;  // sum over this wave's 16 rows
  }
  __syncthreads();

  // Reduce across the 4 waves, apply mean, scatter
  if (tid < H_) {
    float tot = partial[0][tid] + partial[1][tid] + partial[2][tid] + partial[3][tid];
    const float denom = (num > 0) ? (float)num : 1.f;
    out[((size_t)b * MAXN2_ + (size_t)aimN) * H_ + tid] = tot / denom;
  }
}

extern "C" void kernel_launch(void* const* d_in, const int* in_sizes, int n_in,
                              void* d_out, int out_size, void* d_ws, size_t ws_size,
                              hipStream_t stream) {
  const float* node = (const float*)d_in[0];
  const float* ent  = (const float*)d_in[1];
  const float* rel  = (const float*)d_in[2];
  const float* wih  = (const float*)d_in[3];
  const float* whh  = (const float*)d_in[4];
  const float* bih  = (const float*)d_in[5];
  const float* bhh  = (const float*)d_in[6];
  const int* aimN   = (const int*)d_in[7];
  const int* aimE   = (const int*)d_in[8];
  const int* nbrN   = (const int*)d_in[9];
  const int* nbrR   = (const int*)d_in[10];
  const int* nbrM   = (const int*)d_in[11];
  float* out = (float*)d_out;

  const int total = B_ * MAXN2_ * H_;      // 16,793,600 floats (div by 4)
  const int n4 = total / 4;
  Agent_copy_embeddings<<<(n4 + 255) / 256, 256, 0, stream>>>(node, out, n4);
  Agent_gru_agg<<<B_ * T_, 128, 0, stream>>>(node, ent, rel, wih, whh, bih, bhh,
                                             aimN, aimE, nbrN, nbrR, nbrM, out);
}